// ConditionEncoder_7722351199081
// MI455X (gfx1250) — compile-verified
//
#include <hip/hip_runtime.h>
#include <hip/hip_bf16.h>

typedef __attribute__((ext_vector_type(2))) float v2f;
typedef __attribute__((ext_vector_type(8))) float v8f;

// -------------------------------------------------------------------------
// 2x2x2 any-pool on the occupancy mask (jax bool = 1 byte/elem)
// -------------------------------------------------------------------------
__global__ void pool_mask_kernel(const unsigned char* __restrict__ in,
                                 unsigned char* __restrict__ out, int Dout) {
    int o = blockIdx.x * blockDim.x + threadIdx.x;
    int n = Dout * Dout * Dout;
    if (o >= n) return;
    int x = o % Dout, y = (o / Dout) % Dout, z = o / (Dout * Dout);
    int Din = Dout * 2;
    unsigned char v = 0;
    for (int dz = 0; dz < 2; ++dz)
        for (int dy = 0; dy < 2; ++dy)
            for (int dx = 0; dx < 2; ++dx)
                v |= in[(((z * 2 + dz) * Din) + (y * 2 + dy)) * Din + (x * 2 + dx)];
    out[o] = v ? 1 : 0;
}

// -------------------------------------------------------------------------
// Pre-conv: 3x3x3, C=1 -> 32, ReLU, mask.  K=27 is too small for WMMA, so
// scalar VALU with the 864-float filter + bias staged in LDS.
// NOTE: tap loop is kept rolled (#pragma unroll 1) — full unroll made clang
// hoist all 27 guarded loads + LDS reads and spill (512 VGPRs, scratch).
// -------------------------------------------------------------------------
__global__ __launch_bounds__(256) void pre_conv_kernel(
    const float* __restrict__ x, const float* __restrict__ w,
    const float* __restrict__ b, const unsigned char* __restrict__ mask,
    float* __restrict__ q0) {
    __shared__ float lw[27 * 32 + 32];
    for (int i = threadIdx.x; i < 27 * 32; i += 256) lw[i] = w[i];
    if (threadIdx.x < 32) lw[27 * 32 + threadIdx.x] = b[threadIdx.x];
    __syncthreads();

    int v = blockIdx.x * 256 + threadIdx.x;          // 64^3 voxels total
    int xk = v & 63, yk = (v >> 6) & 63, zk = v >> 12;

    float acc[32];
#pragma unroll
    for (int c = 0; c < 32; ++c) acc[c] = lw[27 * 32 + c];

#pragma unroll 1
    for (int t = 0; t < 27; ++t) {
        int dz = t / 9, dy = (t / 3) % 3, dx = t % 3;
        int iz = zk + dz - 1, iy = yk + dy - 1, ix = xk + dx - 1;
        bool ok = (unsigned)iz < 64u && (unsigned)iy < 64u && (unsigned)ix < 64u;
        float xv = ok ? x[(iz * 64 + iy) * 64 + ix] : 0.f;
#pragma unroll
        for (int c = 0; c < 32; ++c) acc[c] = fmaf(xv, lw[t * 32 + c], acc[c]);
    }
    float mv = mask[v] ? 1.f : 0.f;
#pragma unroll
    for (int c = 0; c < 32; ++c)
        q0[v * 32 + c] = fmaxf(acc[c], 0.f) * mv;
}

// -------------------------------------------------------------------------
// Implicit-GEMM 3D conv via V_WMMA_F32_16X16X4_F32 (exact fp32 semantics).
//   M tile = 16 output voxels (linear), N tile = 16*NT output channels,
//   K = KS^3 * CIN consumed 4 at a time (never straddles a filter tap).
// All geometry is compile-time: every load inside the unrolled ks*NT group
// is base + 32-bit voffset + constant immediate (GVS addressing), so there
// is no per-load address arithmetic and register pressure stays low.
// Fragment layouts per CDNA5 ISA 7.12.2 (wave32):
//   A  lane l, vgpr v : m = l&15,  k = 2*(l>>4) + v
//   B  lane l, vgpr v : n = l&15,  k = 2*(l>>4) + v
//   C/D lane l, vgpr v: n = l&15,  m = 8*(l>>4) + v
// One wave per block -> EXEC is all-ones at every WMMA.
// -------------------------------------------------------------------------
template <int STRIDE, int KS, int CIN, int COUT, int LDD, int NT, bool RELU>
__global__ __launch_bounds__(32, 4) void conv_wmma_kernel(
    const float* __restrict__ in, const float* __restrict__ w,
    const float* __restrict__ bias, const unsigned char* __restrict__ mask,
    float* __restrict__ out) {
    constexpr int Dout = 1 << LDD;
    constexpr int Din  = Dout * STRIDE;
    constexpr int pad  = (KS - 1) / 2;

    const int lane  = threadIdx.x;
    const int half  = lane >> 4;           // 0 or 1
    const int l15   = lane & 15;
    const int mbase = blockIdx.x * 16;
    const int nbase = blockIdx.y * (16 * NT);

    // output voxel whose A-row this lane supplies
    const int mv = mbase + l15;
    const int ox = mv & (Dout - 1);
    const int oy = (mv >> LDD) & (Dout - 1);
    const int oz = mv >> (2 * LDD);

    const int koff = half * 2;             // which K pair this lane holds

    v8f acc[NT];
#pragma unroll
    for (int nt = 0; nt < NT; ++nt) {
        v8f z = {0.f, 0.f, 0.f, 0.f, 0.f, 0.f, 0.f, 0.f};
        acc[nt] = z;
    }

    for (int t = 0; t < KS * KS * KS; ++t) {
        const int dz = t / (KS * KS), dy = (t / KS) % KS, dx = t % KS;
        const int iz = oz * STRIDE + dz - pad;
        const int iy = oy * STRIDE + dy - pad;
        const int ix = ox * STRIDE + dx - pad;
        const bool ok = (unsigned)iz < (unsigned)Din &&
                        (unsigned)iy < (unsigned)Din &&
                        (unsigned)ix < (unsigned)Din;
        // one 32-bit offset each for A and B this tap; all inner-loop loads
        // address base + offset + compile-time immediate
        const int aoff = ((iz * Din + iy) * Din + ix) * CIN + koff;
        const int boff = (t * CIN + koff) * COUT + nbase + l15;
        __builtin_prefetch(w + boff, 0, 1);      // global_prefetch_b8

#pragma unroll 4
        for (int ks = 0; ks < CIN / 4; ++ks) {
            v2f a;
            if (ok) a = *reinterpret_cast<const v2f*>(in + aoff + ks * 4);
            else    { a.x = 0.f; a.y = 0.f; }
#pragma unroll
            for (int nt = 0; nt < NT; ++nt) {
                v2f b;
                b.x = w[boff + ks * 4 * COUT + nt * 16];
                b.y = w[boff + ks * 4 * COUT + nt * 16 + COUT];
                // (neg_a, A, neg_b, B, c_mod, C, reuse_a, reuse_b)
                acc[nt] = __builtin_amdgcn_wmma_f32_16x16x4_f32(
                              false, a, false, b, (short)0, acc[nt], false, false);
            }
        }
    }

    // epilogue: + bias, optional ReLU, occupancy mask, store
    float mrowf[8];
#pragma unroll
    for (int v = 0; v < 8; ++v)
        mrowf[v] = mask[mbase + half * 8 + v] ? 1.f : 0.f;

#pragma unroll
    for (int nt = 0; nt < NT; ++nt) {
        const int n  = nbase + nt * 16 + l15;
        const float bn = bias[n];
#pragma unroll
        for (int v = 0; v < 8; ++v) {
            const int mrow = mbase + half * 8 + v;
            float val = acc[nt][v] + bn;
            if (RELU) val = fmaxf(val, 0.f);
            out[mrow * COUT + n] = val * mrowf[v];
        }
    }
}

// -------------------------------------------------------------------------
extern "C" void kernel_launch(void* const* d_in, const int* in_sizes, int n_in,
                              void* d_out, int out_size, void* d_ws, size_t ws_size,
                              hipStream_t stream) {
    (void)in_sizes; (void)n_in; (void)out_size; (void)ws_size;

    const float* x             = (const float*)d_in[0];
    const unsigned char* mask0 = (const unsigned char*)d_in[1];  // 64^3 bool
    const float* pre_w = (const float*)d_in[2];
    const float* pre_b = (const float*)d_in[3];
    const float* dw0   = (const float*)d_in[4];
    const float* db0   = (const float*)d_in[5];
    const float* dw1   = (const float*)d_in[6];
    const float* db1   = (const float*)d_in[7];
    const float* p1w0  = (const float*)d_in[8];
    const float* p1b0  = (const float*)d_in[9];
    const float* p2w0  = (const float*)d_in[10];
    const float* p2b0  = (const float*)d_in[11];
    const float* p3w0  = (const float*)d_in[12];
    const float* p3b0  = (const float*)d_in[13];
    const float* p1w1  = (const float*)d_in[14];
    const float* p1b1  = (const float*)d_in[15];
    const float* p2w1  = (const float*)d_in[16];
    const float* p2b1  = (const float*)d_in[17];
    const float* p3w1  = (const float*)d_in[18];
    const float* p3b1  = (const float*)d_in[19];

    // outputs concatenated in return order: (q2, out0, out1)
    float* outq = (float*)d_out;                        // 16^3 * 96
    float* out0 = outq + (size_t)16 * 16 * 16 * 96;     // 32^3 * 128
    float* out1 = out0 + (size_t)32 * 32 * 32 * 128;    // 16^3 * 64

    // workspace carve-out
    char* ws = (char*)d_ws;
    float* q0 = (float*)ws; ws += (size_t)64 * 64 * 64 * 32 * 4;  // 33.6 MB
    float* q1 = (float*)ws; ws += (size_t)32 * 32 * 32 * 64 * 4;  //  8.4 MB
    float* pa = (float*)ws; ws += (size_t)32 * 32 * 32 * 64 * 4;  //  8.4 MB
    float* pb = (float*)ws; ws += (size_t)32 * 32 * 32 * 64 * 4;  //  8.4 MB
    unsigned char* m1 = (unsigned char*)ws; ws += 32 * 32 * 32;
    unsigned char* m2 = (unsigned char*)ws; ws += 16 * 16 * 16;

    // masks
    pool_mask_kernel<<<(32768 + 255) / 256, 256, 0, stream>>>(mask0, m1, 32);
    pool_mask_kernel<<<(4096  + 255) / 256, 256, 0, stream>>>(m1,    m2, 16);

    // pre-conv 1->32 @64^3
    pre_conv_kernel<<<262144 / 256, 256, 0, stream>>>(x, pre_w, pre_b, mask0, q0);

    // level 0 @32^3  (32768/16 = 2048 M-tiles, LDD = 5)
    conv_wmma_kernel<2, 3, 32,  64, 5, 4, false><<<dim3(2048, 1), 32, 0, stream>>>(q0, dw0,  db0,  m1, q1);
    conv_wmma_kernel<1, 3, 64,  64, 5, 4, true ><<<dim3(2048, 1), 32, 0, stream>>>(q1, p1w0, p1b0, m1, pa);
    conv_wmma_kernel<1, 1, 64,  64, 5, 4, true ><<<dim3(2048, 1), 32, 0, stream>>>(pa, p2w0, p2b0, m1, pb);
    conv_wmma_kernel<1, 3, 64, 128, 5, 4, false><<<dim3(2048, 2), 32, 0, stream>>>(pb, p3w0, p3b0, m1, out0);

    // level 1 @16^3  (4096/16 = 256 M-tiles, LDD = 4); q2 goes straight to d_out
    conv_wmma_kernel<2, 3, 64,  96, 4, 3, false><<<dim3(256, 2), 32, 0, stream>>>(q1,   dw1,  db1,  m2, outq);
    conv_wmma_kernel<1, 3, 96,  32, 4, 2, true ><<<dim3(256, 1), 32, 0, stream>>>(outq, p1w1, p1b1, m2, pa);
    conv_wmma_kernel<1, 1, 32,  32, 4, 2, true ><<<dim3(256, 1), 32, 0, stream>>>(pa,   p2w1, p2b1, m2, pb);
    conv_wmma_kernel<1, 3, 32,  64, 4, 4, false><<<dim3(256, 1), 32, 0, stream>>>(pb,   p3w1, p3b1, m2, out1);
}